// GATLayer_7060926235080
// MI455X (gfx1250) — compile-verified
//
#include <hip/hip_runtime.h>
#include <math.h>

#define N_NODES   100000
#define N_EDGES   1600000
#define IN_F      128
#define OUT_FEAT  32
#define HEADS     4
#define NEG_SLOPE 0.2f

typedef __attribute__((ext_vector_type(2))) float v2f;
typedef __attribute__((ext_vector_type(8))) float v8f;

// ---------- float atomic max via int/uint ordering trick ----------
__device__ __forceinline__ void atomicMaxFloat(float* addr, float val) {
  if (val >= 0.0f)
    atomicMax((int*)addr, __float_as_int(val));
  else
    atomicMin((unsigned int*)addr, __float_as_uint(val));
}

// ---------- 0: init ----------
__global__ __launch_bounds__(256) void init_kernel(float* __restrict__ out,
                                                   float* __restrict__ ns,
                                                   float* __restrict__ hmax,
                                                   float* __restrict__ hsum) {
  int tid = blockIdx.x * 256 + threadIdx.x;
  if (tid < N_NODES * IN_F) out[tid] = 0.0f;
  if (tid < N_NODES * HEADS) ns[tid] = -__builtin_inff();
  if (tid < HEADS) { hmax[tid] = -__builtin_inff(); hsum[tid] = 0.0f; }
}

// ---------- 1: Wh = h @ W with V_WMMA_F32_16X16X4_F32 ----------
// 256 threads = 8 waves; block computes a 16-row x 128-col tile.
// Wave w owns the 16-column slice [16w, 16w+16).
__global__ __launch_bounds__(256) void gemm_wmma_kernel(const float* __restrict__ h,
                                                        const float* __restrict__ W,
                                                        float* __restrict__ Wh) {
  __shared__ float As[16 * 132];   // pitch 132 -> conflict-free strided reads
  const int row0 = blockIdx.x * 16;
  const int t = threadIdx.x;

  // Stage A tile (16x128), zero-pad out-of-range rows (keeps EXEC all-1 for WMMA).
  for (int i = t; i < 16 * IN_F; i += 256) {
    int r = i >> 7, c = i & 127;
    float v = 0.0f;
    if (row0 + r < N_NODES) v = h[(size_t)(row0 + r) * IN_F + c];
    As[r * 132 + c] = v;
  }
  __syncthreads();

  const int wave = t >> 5;
  const int lane = t & 31;
  const int col0 = wave * 16;
  const int half = lane >> 4;   // 0: lanes 0-15 (K, K+1); 1: lanes 16-31 (K+2, K+3)
  const int l16  = lane & 15;

  v8f acc = {};
  for (int k = 0; k < IN_F; k += 4) {
    const int kk = k + 2 * half;
    v2f a, b;
    // A 16x4 f32 layout: lane<16 holds row l16 (K=kk,kk+1); lane>=16 same row (K=kk+2,kk+3)
    a.x = As[l16 * 132 + kk];
    a.y = As[l16 * 132 + kk + 1];
    // B 4x16 f32 layout: VGPR0 = row K (lanes0-15) / K+2 (lanes16-31); VGPR1 = K+1 / K+3
    b.x = W[(size_t)kk * IN_F + col0 + l16];
    b.y = W[(size_t)(kk + 1) * IN_F + col0 + l16];
    acc = __builtin_amdgcn_wmma_f32_16x16x4_f32(false, a, false, b,
                                                (short)0, acc, false, false);
  }

  // C/D layout: VGPR j -> M = j (lanes 0-15) or j+8 (lanes 16-31), N = lane%16
  #pragma unroll
  for (int j = 0; j < 8; ++j) {
    int r = j + 8 * half;
    if (row0 + r < N_NODES)
      Wh[(size_t)(row0 + r) * IN_F + col0 + l16] = acc[j];
  }
}

// ---------- 2: per-(node,head) attention logits ----------
__global__ __launch_bounds__(256) void node_logits_kernel(const float* __restrict__ Wh,
                                                          const float* __restrict__ a,
                                                          float* __restrict__ s_src,
                                                          float* __restrict__ s_dst) {
  int tid = blockIdx.x * 256 + threadIdx.x;
  if (tid >= N_NODES * HEADS) return;
  int n = tid >> 2, hd = tid & 3;
  const float* w  = Wh + (size_t)n * IN_F + hd * OUT_FEAT;
  const float* as = a + hd * (2 * OUT_FEAT);
  const float* ad = as + OUT_FEAT;
  float ss = 0.0f, sd = 0.0f;
  #pragma unroll
  for (int j = 0; j < OUT_FEAT; ++j) { float v = w[j]; ss += v * as[j]; sd += v * ad[j]; }
  s_src[tid] = ss;
  s_dst[tid] = sd;
}

// ---------- 3: per-(edge,head) leaky-relu score + segment max into node_scores ----------
__global__ __launch_bounds__(256) void edge_scores_kernel(const int* __restrict__ ei,
                                                          const float* __restrict__ s_src,
                                                          const float* __restrict__ s_dst,
                                                          float* __restrict__ ns) {
  int tid = blockIdx.x * 256 + threadIdx.x;
  if (tid >= N_EDGES * HEADS) return;
  int e = tid >> 2, hd = tid & 3;
  int s = ei[e];
  int d = ei[N_EDGES + e];
  float v = s_src[s * HEADS + hd] + s_dst[d * HEADS + hd];
  v = (v > 0.0f) ? v : NEG_SLOPE * v;
  atomicMaxFloat(&ns[d * HEADS + hd], v);
}

// ---------- 4: per-head max over all nodes ----------
#define NODES_PER_BLOCK 512
__global__ __launch_bounds__(256) void head_max_kernel(const float* __restrict__ ns,
                                                       float* __restrict__ hmax) {
  __shared__ float sm[256];
  int t = threadIdx.x;
  int hd = t & 3;
  int nstart = blockIdx.x * NODES_PER_BLOCK;
  int nend = nstart + NODES_PER_BLOCK; if (nend > N_NODES) nend = N_NODES;
  float m = -__builtin_inff();
  for (int n = nstart + (t >> 2); n < nend; n += 64)
    m = fmaxf(m, ns[n * HEADS + hd]);
  sm[t] = m;
  __syncthreads();
  for (int s = 128; s >= 4; s >>= 1) {   // s always multiple of 4 -> same head combines
    if (t < s) sm[t] = fmaxf(sm[t], sm[t + s]);
    __syncthreads();
  }
  if (t < HEADS) atomicMaxFloat(&hmax[t], sm[t]);
}

// ---------- 5: per-head sum of exp ----------
__global__ __launch_bounds__(256) void head_sum_kernel(const float* __restrict__ ns,
                                                       const float* __restrict__ hmax,
                                                       float* __restrict__ hsum) {
  __shared__ float sm[256];
  int t = threadIdx.x;
  int hd = t & 3;
  float mx = hmax[hd];
  int nstart = blockIdx.x * NODES_PER_BLOCK;
  int nend = nstart + NODES_PER_BLOCK; if (nend > N_NODES) nend = N_NODES;
  float acc = 0.0f;
  for (int n = nstart + (t >> 2); n < nend; n += 64) {
    float v = ns[n * HEADS + hd];
    acc += __expf(v - mx);   // exp(-inf - mx) = 0 for empty segments
  }
  sm[t] = acc;
  __syncthreads();
  for (int s = 128; s >= 4; s >>= 1) {
    if (t < s) sm[t] += sm[t + s];
    __syncthreads();
  }
  if (t < HEADS) atomicAdd(&hsum[t], sm[t]);
}

// ---------- 6: per-node attention weight ----------
__global__ __launch_bounds__(256) void attn_kernel(const float* __restrict__ ns,
                                                   const float* __restrict__ hmax,
                                                   const float* __restrict__ hsum,
                                                   float* __restrict__ attn) {
  int tid = blockIdx.x * 256 + threadIdx.x;
  if (tid >= N_NODES * HEADS) return;
  int hd = tid & 3;
  attn[tid] = __expf(ns[tid] - hmax[hd]) / hsum[hd];
}

// ---------- 7: edge aggregation: out[dst] += Wh[src] ----------
// 256 threads = 8 edges/block, each thread handles 4 consecutive features (float4 gather).
__global__ __launch_bounds__(256) void aggregate_kernel(const float* __restrict__ Wh,
                                                        const int* __restrict__ ei,
                                                        float* __restrict__ out) {
  int t = threadIdx.x;
  int e = blockIdx.x * 8 + (t >> 5);
  if (e >= N_EDGES) return;
  int s = ei[e];
  int d = ei[N_EDGES + e];
  int f = (t & 31) * 4;
  const float4 v = *(const float4*)(Wh + (size_t)s * IN_F + f);
  float* o = out + (size_t)d * IN_F + f;
  atomicAdd(o + 0, v.x);
  atomicAdd(o + 1, v.y);
  atomicAdd(o + 2, v.z);
  atomicAdd(o + 3, v.w);
}

// ---------- 8: scale by attn(dst-node) : out[n,h,:] *= attn[n,h] ----------
__global__ __launch_bounds__(256) void scale_kernel(const float* __restrict__ attn,
                                                    float* __restrict__ out) {
  int tid = blockIdx.x * 256 + threadIdx.x;
  if (tid >= N_NODES * IN_F) return;
  int n = tid >> 7;
  int hd = (tid & 127) >> 5;
  out[tid] *= attn[n * HEADS + hd];
}

extern "C" void kernel_launch(void* const* d_in, const int* in_sizes, int n_in,
                              void* d_out, int out_size, void* d_ws, size_t ws_size,
                              hipStream_t stream) {
  const float* h  = (const float*)d_in[0];
  const int*   ei = (const int*)d_in[1];
  const float* W  = (const float*)d_in[2];
  const float* a  = (const float*)d_in[3];
  float* out = (float*)d_out;

  // workspace layout (floats)
  float* Wh    = (float*)d_ws;                         // N*128
  float* s_src = Wh    + (size_t)N_NODES * IN_F;       // N*4
  float* s_dst = s_src + (size_t)N_NODES * HEADS;      // N*4
  float* ns    = s_dst + (size_t)N_NODES * HEADS;      // N*4
  float* attn  = ns    + (size_t)N_NODES * HEADS;      // N*4
  float* hmax  = attn  + (size_t)N_NODES * HEADS;      // 4
  float* hsum  = hmax  + HEADS;                        // 4

  const int nh  = N_NODES * HEADS;
  const int eh  = N_EDGES * HEADS;
  const int nf  = N_NODES * IN_F;
  const int red_blocks = (N_NODES + NODES_PER_BLOCK - 1) / NODES_PER_BLOCK;

  init_kernel<<<(nf + 255) / 256, 256, 0, stream>>>(out, ns, hmax, hsum);
  gemm_wmma_kernel<<<(N_NODES + 15) / 16, 256, 0, stream>>>(h, W, Wh);
  node_logits_kernel<<<(nh + 255) / 256, 256, 0, stream>>>(Wh, a, s_src, s_dst);
  edge_scores_kernel<<<(eh + 255) / 256, 256, 0, stream>>>(ei, s_src, s_dst, ns);
  head_max_kernel<<<red_blocks, 256, 0, stream>>>(ns, hmax);
  head_sum_kernel<<<red_blocks, 256, 0, stream>>>(ns, hmax, hsum);
  attn_kernel<<<(nh + 255) / 256, 256, 0, stream>>>(ns, hmax, hsum, attn);
  aggregate_kernel<<<(N_EDGES + 7) / 8, 256, 0, stream>>>(Wh, ei, out);
  scale_kernel<<<(nf + 255) / 256, 256, 0, stream>>>(attn, out);
}